// KANLayer_10703058502229
// MI455X (gfx1250) — compile-verified
//
#include <hip/hip_runtime.h>
#include <hip/hip_bf16.h>

typedef __attribute__((ext_vector_type(2))) float v2f;
typedef __attribute__((ext_vector_type(8))) float v8f;

#define B_    4096
#define DIN_  768
#define DOUT_ 512
#define Q_    64
#define H1_   32
#define H2_   64
#define EPS_  1e-5f

// ---------------------------------------------------------------------------
// K1: x_proj[B,DOUT] = x[B,DIN] @ wp[DIN,DOUT] + bp
// One wave per block; each wave owns a 16-row x 64-col tile (4 WMMA N-tiles),
// K loop over DIN in steps of 4 using V_WMMA_F32_16X16X4_F32.
// A layout: lane holds A[lane&15][2*(lane>>4)+i] (i=0,1)
// B layout: lane holds B[2*(lane>>4)+i][lane&15]
// C layout: lane holds C[i + 8*(lane>>4)][lane&15] for i=0..7
// ---------------------------------------------------------------------------
__global__ __launch_bounds__(32)
void k_gemm_xproj(const float* __restrict__ x, const float* __restrict__ wp,
                  const float* __restrict__ bp, float* __restrict__ xp) {
  const int lane = threadIdx.x;
  const int r    = lane & 15;
  const int kh   = lane >> 4;              // 0 or 1
  const int row0 = blockIdx.y * 16;
  const int col0 = blockIdx.x * 64;

  v8f acc0 = 0.0f, acc1 = 0.0f, acc2 = 0.0f, acc3 = 0.0f;

  const float* __restrict__ arow = x + (size_t)(row0 + r) * DIN_;
  for (int k = 0; k < DIN_; k += 4) {
    const int ka = k + kh * 2;
    v2f a = *(const v2f*)(arow + ka);      // 8B-aligned (ka even)

    v2f b0, b1, b2, b3;
    const float* __restrict__ w0 = wp + (size_t)ka * DOUT_ + col0 + r;
    const float* __restrict__ w1 = w0 + DOUT_;
    b0.x = w0[0];  b0.y = w1[0];
    b1.x = w0[16]; b1.y = w1[16];
    b2.x = w0[32]; b2.y = w1[32];
    b3.x = w0[48]; b3.y = w1[48];

    acc0 = __builtin_amdgcn_wmma_f32_16x16x4_f32(false, a, false, b0, (short)0, acc0, false, false);
    acc1 = __builtin_amdgcn_wmma_f32_16x16x4_f32(false, a, false, b1, (short)0, acc1, false, false);
    acc2 = __builtin_amdgcn_wmma_f32_16x16x4_f32(false, a, false, b2, (short)0, acc2, false, false);
    acc3 = __builtin_amdgcn_wmma_f32_16x16x4_f32(false, a, false, b3, (short)0, acc3, false, false);
  }

  v8f accs[4] = {acc0, acc1, acc2, acc3};
#pragma unroll
  for (int t = 0; t < 4; ++t) {
    const int col  = col0 + t * 16 + r;
    const float bv = bp[col];
#pragma unroll
    for (int i = 0; i < 8; ++i) {
      const int row = row0 + i + 8 * kh;
      xp[(size_t)row * DOUT_ + col] = accs[t][i] + bv;
    }
  }
}

// ---------------------------------------------------------------------------
// K2: per-row fused inner+outer reduction (sum commutes with shared linear):
//   H[h]  = sum_d relu(xp[b,d]*wi1[h] + bi1[h])          (d < 512, h < 32)
//   u[q]  = sum_h H[h]*wi2[h,q] + DOUT*bi2[q]            (q < 64)
//   G[b,h2] = sum_q relu(u[q]*wo1[h2] + bo1[h2])         (h2 < 64)
// One block (256 threads) per row b.
// ---------------------------------------------------------------------------
__global__ __launch_bounds__(256)
void k_inner_outer(const float* __restrict__ xp,
                   const float* __restrict__ wi1, const float* __restrict__ bi1,
                   const float* __restrict__ wi2, const float* __restrict__ bi2,
                   const float* __restrict__ wo1, const float* __restrict__ bo1,
                   float* __restrict__ G) {
  __shared__ float sH[8][H1_];
  __shared__ float sU[Q_];

  const int b     = blockIdx.x;
  const int tid   = threadIdx.x;
  const int h     = tid & (H1_ - 1);   // 0..31
  const int chunk = tid >> 5;          // 0..7, each covers 64 d-values

  const float w1  = wi1[h];
  const float bb1 = bi1[h];
  const float* __restrict__ xrow = xp + (size_t)b * DOUT_;

  float s = 0.0f;
  const int d0 = chunk * 64;
#pragma unroll 4
  for (int d = d0; d < d0 + 64; ++d) {
    s += fmaxf(fmaf(xrow[d], w1, bb1), 0.0f);
  }
  sH[chunk][h] = s;
  __syncthreads();

  if (tid < Q_) {
    float u = (float)DOUT_ * bi2[tid];
#pragma unroll
    for (int hh = 0; hh < H1_; ++hh) {
      float Hh = 0.0f;
#pragma unroll
      for (int c = 0; c < 8; ++c) Hh += sH[c][hh];
      u = fmaf(Hh, wi2[hh * Q_ + tid], u);
    }
    sU[tid] = u;
  }
  __syncthreads();

  if (tid < H2_) {
    const float w  = wo1[tid];
    const float bb = bo1[tid];
    float g = 0.0f;
#pragma unroll 8
    for (int q = 0; q < Q_; ++q) g += fmaxf(fmaf(sU[q], w, bb), 0.0f);
    G[(size_t)b * H2_ + tid] = g;
  }
}

// ---------------------------------------------------------------------------
// K3: summed[B,DOUT] = G[B,64] @ wo2[64,DOUT] + Q*bo2, then row LayerNorm.
// 16 rows per block, 8 waves; wave w owns N-tiles [w*64, w*64+64), K=64.
// Results staged in LDS (16x512 f32 = 32KB) for the LN statistics.
// ---------------------------------------------------------------------------
__global__ __launch_bounds__(256)
void k_outer_gemm_ln(const float* __restrict__ G, const float* __restrict__ wo2,
                     const float* __restrict__ bo2,
                     const float* __restrict__ gamma, const float* __restrict__ beta,
                     float* __restrict__ out) {
  __shared__ float sS[16][DOUT_];      // 32 KB
  __shared__ float sRed[16][16][2];
  __shared__ float sMean[16];
  __shared__ float sRstd[16];

  const int tid  = threadIdx.x;
  const int lane = tid & 31;
  const int wave = tid >> 5;           // 0..7
  const int r    = lane & 15;
  const int kh   = lane >> 4;
  const int row0 = blockIdx.x * 16;
  const int colw = wave * 64;

  v8f acc0 = 0.0f, acc1 = 0.0f, acc2 = 0.0f, acc3 = 0.0f;

  const float* __restrict__ arow = G + (size_t)(row0 + r) * H2_;
  for (int k = 0; k < H2_; k += 4) {
    const int ka = k + kh * 2;
    v2f a = *(const v2f*)(arow + ka);

    const float* __restrict__ w0 = wo2 + (size_t)ka * DOUT_ + colw + r;
    const float* __restrict__ w1 = w0 + DOUT_;
    v2f b0, b1, b2, b3;
    b0.x = w0[0];  b0.y = w1[0];
    b1.x = w0[16]; b1.y = w1[16];
    b2.x = w0[32]; b2.y = w1[32];
    b3.x = w0[48]; b3.y = w1[48];

    acc0 = __builtin_amdgcn_wmma_f32_16x16x4_f32(false, a, false, b0, (short)0, acc0, false, false);
    acc1 = __builtin_amdgcn_wmma_f32_16x16x4_f32(false, a, false, b1, (short)0, acc1, false, false);
    acc2 = __builtin_amdgcn_wmma_f32_16x16x4_f32(false, a, false, b2, (short)0, acc2, false, false);
    acc3 = __builtin_amdgcn_wmma_f32_16x16x4_f32(false, a, false, b3, (short)0, acc3, false, false);
  }

  v8f accs[4] = {acc0, acc1, acc2, acc3};
#pragma unroll
  for (int t = 0; t < 4; ++t) {
    const int col  = colw + t * 16 + r;
    const float bv = (float)Q_ * bo2[col];
#pragma unroll
    for (int i = 0; i < 8; ++i) {
      sS[i + 8 * kh][col] = accs[t][i] + bv;
    }
  }
  __syncthreads();

  // Row statistics: 16 threads per row, each sums 32 of 512 columns.
  {
    const int row = tid >> 4;
    const int sub = tid & 15;
    float s = 0.0f, ss = 0.0f;
    const int c0 = sub * 32;
#pragma unroll 8
    for (int c = c0; c < c0 + 32; ++c) {
      const float v = sS[row][c];
      s += v;
      ss = fmaf(v, v, ss);
    }
    sRed[row][sub][0] = s;
    sRed[row][sub][1] = ss;
  }
  __syncthreads();

  if (tid < 16) {
    float s = 0.0f, ss = 0.0f;
#pragma unroll
    for (int i = 0; i < 16; ++i) { s += sRed[tid][i][0]; ss += sRed[tid][i][1]; }
    const float mean = s * (1.0f / DOUT_);
    const float var  = ss * (1.0f / DOUT_) - mean * mean;
    sMean[tid] = mean;
    sRstd[tid] = rsqrtf(var + EPS_);
  }
  __syncthreads();

  for (int idx = tid; idx < 16 * DOUT_; idx += 256) {
    const int row = idx >> 9;            // / 512
    const int col = idx & (DOUT_ - 1);
    const float v = (sS[row][col] - sMean[row]) * sRstd[row];
    out[(size_t)(row0 + row) * DOUT_ + col] = fmaf(v, gamma[col], beta[col]);
  }
}

// ---------------------------------------------------------------------------
// Host-side launcher. Input order (setup_inputs):
// 0:x 1:wp 2:bp 3:wi1 4:bi1 5:wi2 6:bi2 7:wo1 8:bo1 9:wo2 10:bo2 11:gamma 12:beta
// ---------------------------------------------------------------------------
extern "C" void kernel_launch(void* const* d_in, const int* in_sizes, int n_in,
                              void* d_out, int out_size, void* d_ws, size_t ws_size,
                              hipStream_t stream) {
  (void)in_sizes; (void)n_in; (void)out_size; (void)ws_size;

  const float* x     = (const float*)d_in[0];
  const float* wp    = (const float*)d_in[1];
  const float* bp    = (const float*)d_in[2];
  const float* wi1   = (const float*)d_in[3];
  const float* bi1   = (const float*)d_in[4];
  const float* wi2   = (const float*)d_in[5];
  const float* bi2   = (const float*)d_in[6];
  const float* wo1   = (const float*)d_in[7];
  const float* bo1   = (const float*)d_in[8];
  const float* wo2   = (const float*)d_in[9];
  const float* bo2   = (const float*)d_in[10];
  const float* gamma = (const float*)d_in[11];
  const float* beta  = (const float*)d_in[12];

  float* xp = (float*)d_ws;                                           // [B, DOUT]  8 MB
  float* G  = (float*)((char*)d_ws + (size_t)B_ * DOUT_ * sizeof(float)); // [B, H2] 1 MB
  float* o  = (float*)d_out;

  k_gemm_xproj<<<dim3(DOUT_ / 64, B_ / 16), 32, 0, stream>>>(x, wp, bp, xp);
  k_inner_outer<<<B_, 256, 0, stream>>>(xp, wi1, bi1, wi2, bi2, wo1, bo1, G);
  k_outer_gemm_ln<<<B_ / 16, 256, 0, stream>>>(G, wo2, bo2, gamma, beta, o);
}